// HybridKernel_65481071397151
// MI455X (gfx1250) — compile-verified
//
#include <hip/hip_runtime.h>
#include <math.h>

// ---------------------------------------------------------------------------
// CDNA5 WMMA (wave32): V_WMMA_F32_16X16X4_F32
//   A: 16x4 f32  -> 2 VGPRs/lane (v2f)   lane = m + 16*(k/2%2), vgpr = k%2
//   B: 4x16 f32  -> 2 VGPRs/lane (v2f)   lane = n + 16*(k/2%2), vgpr = k%2
//   C/D: 16x16 f32 -> 8 VGPRs/lane (v8f) M = v + 8*(lane/16), N = lane%16
// ---------------------------------------------------------------------------
typedef float v2f __attribute__((ext_vector_type(2)));
typedef float v8f __attribute__((ext_vector_type(8)));

#define N_WIRES 4
#define DIM 16
#define N_OPS 30
#define N_STEPS (N_OPS + 2 * N_WIRES)   // 30 random gates + RX/RY per wire

// ---------------------------------------------------------------------------
// Kernel A: single wave32, latency-critical serial path.
// Compose U = g37 ... g0 (16x16 complex) with f32 WMMA, then
// qk = | r^T U v | with v,r the product-state vectors from x[0,:], y[0,:].
// All trig is done ONCE, in parallel across lanes, with HW v_sin/v_cos.
// ---------------------------------------------------------------------------
__global__ __launch_bounds__(32)
void hk_compose_qk(const float* __restrict__ x,          // (B,4), use row 0
                   const float* __restrict__ y,          // (B,4), use row 0
                   const float* __restrict__ rand_angles,// (30,)
                   const float* __restrict__ rx_param,   // (1,)
                   const float* __restrict__ ry_param,   // (1,)
                   float* __restrict__ qk_out) {
    __shared__ float Sre[2][256];
    __shared__ float Sim[2][256];
    __shared__ float cA[32], sA[32];    // half-angle cos/sin per gate param
    __shared__ float vv[16], rr[16];

    const int lane = threadIdx.x;      // 0..31
    const int m    = lane & 15;        // row (A) / col (B,D)
    const int grp  = lane >> 4;        // 0 or 1 (K/M half selector)

    // --- phase 0: all gate trig in ONE parallel shot (lanes 0..31) ---
    {
        float ang;
        if (lane < N_OPS)       ang = rand_angles[lane];
        else if (lane == 30)    ang = rx_param[0];
        else                    ang = ry_param[0];
        float h = 0.5f * ang;
        cA[lane] = __cosf(h);
        sA[lane] = __sinf(h);
    }

    // --- init U = I in buffer 0 (8 entries per lane) ---
    #pragma unroll
    for (int t = 0; t < 8; ++t) {
        int e = lane + 32 * t;         // flat index row*16+col
        Sre[0][e] = ((e >> 4) == (e & 15)) ? 1.0f : 0.0f;
        Sim[0][e] = 0.0f;
    }
    __syncthreads();

    int cur = 0;
    for (int step = 0; step < N_STEPS; ++step) {
        const int nxt = cur ^ 1;

        // ---- uniform gate selection (no trig here) ----
        int gkind, idx, w;             // gkind: 0 RX, 1 RY, 2 RZ, 3 CNOT
        if (step < N_OPS) {
            gkind = step & 3;          // N_WIRES==4 -> type == wire
            idx   = step;
            w     = gkind;             // wire (unused for CNOT)
        } else {
            int k = step - N_OPS;
            gkind = k & 1;             // even: RX, odd: RY
            idx   = N_OPS + gkind;     // 30 -> rx_param, 31 -> ry_param
            w     = k >> 1;
        }

        if (gkind == 3) {
            // ---- CNOT(wire3 -> wire0) is a row permutation: no matmul ----
            // f(i) flips bit3 (target wire 0) if bit0 (control wire 3) set
            #pragma unroll
            for (int t = 0; t < 8; ++t) {
                int e = lane + 32 * t;
                int i = e >> 4, j = e & 15;
                int fi = (i & 1) ? (i ^ 8) : i;
                Sre[nxt][e] = Sre[cur][fi * 16 + j];
                Sim[nxt][e] = Sim[cur][fi * 16 + j];
            }
            __syncthreads();
            cur = nxt;
            continue;
        }

        const float ch = cA[idx];
        const float sh = sA[idx];
        // 2x2 gate entries (complex), built branch-free per kind
        float g00r = ch, g11r = ch;
        float g01r = 0.f, g10r = 0.f;
        float g00i = 0.f, g01i = 0.f, g10i = 0.f, g11i = 0.f;
        bool has_imag = false;
        if (gkind == 0) {              // RX: [[c,-is],[-is,c]]
            g01i = -sh; g10i = -sh; has_imag = true;
        } else if (gkind == 1) {       // RY: [[c,-s],[s,c]]
            g01r = -sh; g10r = sh;
        } else {                       // RZ: diag(e^{-ih}, e^{ih})
            g00i = -sh; g11i = sh; has_imag = true;
        }
        const int p     = 3 - w;             // bit position of wire w
        const int omask = 15 ^ (1 << p);     // "other bits" mask

        // ---- U_new = G * U : 4 K-chunks, 2 or 4 real f32 WMMAs each ----
        v8f accR = {}; v8f accI = {};
        #pragma unroll
        for (int kc = 0; kc < 4; ++kc) {
            v2f aR, aI, aNI, bR, bI;
            #pragma unroll
            for (int s = 0; s < 2; ++s) {
                int k = 4 * kc + s + 2 * grp;    // A-layout K for lane/slot
                bool same = ((m & omask) == (k & omask));
                int bm = (m >> p) & 1;
                int bk = (k >> p) & 1;
                float er = bm ? (bk ? g11r : g10r) : (bk ? g01r : g00r);
                float ei = bm ? (bk ? g11i : g10i) : (bk ? g01i : g00i);
                aR[s]  = same ? er : 0.0f;
                float gi = same ? ei : 0.0f;
                aI[s]  = gi;
                aNI[s] = -gi;
                // B-layout gather of U (row k, col n=m) from LDS
                bR[s] = Sre[cur][k * 16 + m];
                bI[s] = Sim[cur][k * 16 + m];
            }
            // Dr += Gr*Ur ; Di += Gr*Ui   (always)
            accR = __builtin_amdgcn_wmma_f32_16x16x4_f32(false, aR, false, bR, (short)0, accR, false, false);
            accI = __builtin_amdgcn_wmma_f32_16x16x4_f32(false, aR, false, bI, (short)0, accI, false, false);
            if (has_imag) {            // uniform branch: EXEC stays all-ones
                // Dr += (-Gi)*Ui ; Di += Gi*Ur
                accR = __builtin_amdgcn_wmma_f32_16x16x4_f32(false, aNI, false, bI, (short)0, accR, false, false);
                accI = __builtin_amdgcn_wmma_f32_16x16x4_f32(false, aI,  false, bR, (short)0, accI, false, false);
            }
        }

        // ---- scatter D (C/D layout) into ping-pong buffer ----
        #pragma unroll
        for (int v = 0; v < 8; ++v) {
            int mm = v + 8 * grp;
            Sre[nxt][mm * 16 + m] = accR[v];
            Sim[nxt][mm * 16 + m] = accI[v];
        }
        __syncthreads();
        cur = nxt;
    }

    // ---- product-state vectors: v from x[0,:], r from y[0,:] ----
    // wire w lives at bit (3-w); amplitude factor = bit ? sin(h) : cos(h)
    if (lane < 16) {
        float pv = 1.0f, pr = 1.0f;
        #pragma unroll
        for (int ww = 0; ww < 4; ++ww) {
            int bit = (lane >> (3 - ww)) & 1;
            float hx = 0.5f * x[ww];
            float hy = 0.5f * y[ww];
            pv *= bit ? __sinf(hx) : __cosf(hx);
            pr *= bit ? __sinf(hy) : __cosf(hy);
        }
        vv[lane] = pv;
        rr[lane] = pr;
    }
    __syncthreads();

    // ---- z = sum_ij r_i * U_ij * v_j  (8 entries per lane) ----
    float zr = 0.0f, zi = 0.0f;
    #pragma unroll
    for (int t = 0; t < 8; ++t) {
        int e = lane + 32 * t;
        int i = e >> 4, j = e & 15;
        float wgt = rr[i] * vv[j];
        zr += wgt * Sre[cur][e];
        zi += wgt * Sim[cur][e];
    }
    // wave32 butterfly reduction
    #pragma unroll
    for (int off = 16; off > 0; off >>= 1) {
        zr += __shfl_xor(zr, off, 32);
        zi += __shfl_xor(zi, off, 32);
    }
    if (lane == 0) qk_out[0] = sqrtf(zr * zr + zi * zi);
}

// ---------------------------------------------------------------------------
// Kernel B: bandwidth-bound streaming RBF. 32B read + 4B write per element,
// 128-bit vector loads. out[i] = 0.5*exp(-||x_i - y_i||^2) + 0.5*qk
// Roofline: 1M * 36B = 37.7 MB @ 23.3 TB/s ~= 1.6 us.
// ---------------------------------------------------------------------------
__global__ __launch_bounds__(256)
void hk_batch(const float4* __restrict__ x,
              const float4* __restrict__ y,
              const float*  __restrict__ qk,
              float* __restrict__ out, int B) {
    int i = blockIdx.x * 256 + threadIdx.x;
    if (i >= B) return;
    float4 a = x[i];
    float4 b = y[i];
    float dx = a.x - b.x;
    float dy = a.y - b.y;
    float dz = a.z - b.z;
    float dw = a.w - b.w;
    float d2 = dx * dx + dy * dy + dz * dz + dw * dw;
    out[i] = 0.5f * __expf(-d2) + 0.5f * qk[0];   // qk[0]: uniform scalar load
}

extern "C" void kernel_launch(void* const* d_in, const int* in_sizes, int n_in,
                              void* d_out, int out_size, void* d_ws, size_t ws_size,
                              hipStream_t stream) {
    const float* x  = (const float*)d_in[0];   // (B,4) f32
    const float* yy = (const float*)d_in[1];   // (B,4) f32
    const float* ra = (const float*)d_in[2];   // (30,) f32
    const float* rx = (const float*)d_in[3];   // (1,)  f32
    const float* ry = (const float*)d_in[4];   // (1,)  f32
    float* out = (float*)d_out;
    float* qk  = (float*)d_ws;                 // scratch scalar

    const int B = in_sizes[0] / 4;

    // 1) tiny single-wave WMMA kernel -> qk scalar in d_ws
    hk_compose_qk<<<1, 32, 0, stream>>>(x, yy, ra, rx, ry, qk);

    // 2) bandwidth-bound streaming kernel over the 1M batch
    int blocks = (B + 255) / 256;
    hk_batch<<<blocks, 256, 0, stream>>>((const float4*)x, (const float4*)yy,
                                         qk, out, B);
}